// Encoder_DGI_6081673691169
// MI455X (gfx1250) — compile-verified
//
#include <hip/hip_runtime.h>
#include <hip/hip_bf16.h>

// GCN (spectral-normed W) + symmetric-norm aggregation + PReLU for MI455X.
// Pipeline (all on `stream`):
//   1. power_iter_kernel : 1/sigma via power iteration on W^T W   -> ws
//   2. fill/count/rsqrt  : dinv[n] = rsqrt(deg_in[n] + 1)          -> ws
//   3. zero_out_kernel   : d_out = 0 (atomic accumulator)
//   4. gemm_wmma_kernel  : h = (x @ W) * (1/sigma)  [v_wmma_f32_16x16x4_f32]
//   5. edge_scatter      : one wave per edge, coalesced float4 gather +
//                          global_atomic_add_f32 scatter
//   6. finalize          : += self-loop term + bias, PReLU

#define N_NODES 100000
#define N_EDGES 1600000
#define NFEAT   256
#define NHID    128

typedef float v2f __attribute__((ext_vector_type(2)));
typedef float v8f __attribute__((ext_vector_type(8)));

// -------------------- 1. spectral norm (power iteration) --------------------
__global__ __launch_bounds__(256) void power_iter_kernel(const float* __restrict__ W,
                                                         float* __restrict__ inv_sigma) {
  __shared__ float v[NHID];
  __shared__ float u[NFEAT];
  __shared__ float red[256];
  const int t = threadIdx.x;
  if (t < NHID) v[t] = 1.0f;
  __syncthreads();
  float nw = 1.0f;
  for (int it = 0; it < 64; ++it) {
    // u = W v : thread t handles row t (NFEAT == 256 == blockDim)
    float acc = 0.0f;
    const float* wr = W + t * NHID;
#pragma unroll 8
    for (int j = 0; j < NHID; ++j) acc += wr[j] * v[j];
    u[t] = acc;
    __syncthreads();
    // w = W^T u : threads t < 128 handle column t
    float wj = 0.0f;
    if (t < NHID) {
#pragma unroll 8
      for (int i = 0; i < NFEAT; ++i) wj += W[i * NHID + t] * u[i];
    }
    red[t] = (t < NHID) ? wj * wj : 0.0f;
    __syncthreads();
    for (int s = 128; s > 0; s >>= 1) {
      if (t < s) red[t] += red[t + s];
      __syncthreads();
    }
    nw = sqrtf(red[0]);            // ||W^T W v||, v was unit norm
    if (t < NHID) v[t] = wj / nw;
    __syncthreads();
  }
  if (t == 0) inv_sigma[0] = rsqrtf(nw);  // sigma = sqrt(lambda_max)
}

// -------------------- 2. degree / dinv --------------------
__global__ void fill_ones_kernel(float* __restrict__ p, int n) {
  int i = blockIdx.x * blockDim.x + threadIdx.x;
  if (i < n) p[i] = 1.0f;  // self-loop contributes 1 to every degree
}

__global__ void deg_count_kernel(const int* __restrict__ col, float* __restrict__ deg, int e) {
  int i = blockIdx.x * blockDim.x + threadIdx.x;
  if (i < e) unsafeAtomicAdd(&deg[col[i]], 1.0f);
}

__global__ void rsqrt_kernel(float* __restrict__ p, int n) {
  int i = blockIdx.x * blockDim.x + threadIdx.x;
  if (i < n) p[i] = rsqrtf(p[i]);  // deg >= 1 always (self-loops)
}

// -------------------- 3. zero accumulator --------------------
__global__ void zero_out_kernel(float* __restrict__ p, long long n) {
  long long i = (long long)blockIdx.x * blockDim.x + threadIdx.x;
  if (i < n) p[i] = 0.0f;
}

// -------------------- 4. GEMM via WMMA f32 16x16x4 --------------------
#define XT_STRIDE 260  // 16-row x tile, padded for conflict-free lane reads
__global__ __launch_bounds__(256) void gemm_wmma_kernel(const float* __restrict__ x,
                                                        const float* __restrict__ W,
                                                        const float* __restrict__ inv_sigma,
                                                        float* __restrict__ h) {
  __shared__ float sx[16 * XT_STRIDE];
  const int t    = threadIdx.x;
  const int wave = t >> 5;          // 8 waves: one 16-col slice of NHID each
  const int lane = t & 31;
  const long long rowBase = (long long)blockIdx.x * 16;

  // stage the contiguous 16x256 tile of x (coalesced)
  const float* xt = x + rowBase * NFEAT;
#pragma unroll
  for (int r = 0; r < 16; ++r) sx[r * XT_STRIDE + t] = xt[r * NFEAT + t];
  __syncthreads();

  // ISA 16x4 f32 A layout: lanes 0-15 -> K=0,1 ; lanes 16-31 -> K=2,3
  const int m    = lane & 15;
  const int ksel = (lane >> 4) << 1;  // 0 or 2
  const int col  = wave * 16 + m;     // B/C column index == lane%16
  const float* srow = sx + m * XT_STRIDE + ksel;

  v8f c = {};
#pragma unroll 8
  for (int kb = 0; kb < NFEAT; kb += 4) {
    v2f a, b;
    a.x = srow[kb];
    a.y = srow[kb + 1];
    b.x = W[(kb + ksel) * NHID + col];
    b.y = W[(kb + ksel + 1) * NHID + col];
    c = __builtin_amdgcn_wmma_f32_16x16x4_f32(false, a, false, b, (short)0, c,
                                              false, false);
  }

  const float is = inv_sigma[0];
  const int mbase = (lane >> 4) * 8;  // C/D layout: lanes16-31 hold rows M+8
#pragma unroll
  for (int r = 0; r < 8; ++r)
    h[(rowBase + mbase + r) * NHID + col] = c[r] * is;
}

// -------------------- 5. edge gather/scatter (one wave per edge) ------------
__global__ __launch_bounds__(256) void edge_scatter_kernel(const int* __restrict__ ei,
                                                           const float* __restrict__ h,
                                                           const float* __restrict__ dinv,
                                                           float* __restrict__ out,
                                                           long long e_count) {
  long long idx = (long long)blockIdx.x * blockDim.x + threadIdx.x;
  long long e = idx >> 5;              // 32 lanes cooperate on one edge
  if (e >= e_count) return;
  int c = (int)(idx & 31);             // float4 chunk: 32*4 = 128 feats
  int src = ei[e];                     // row (source)
  int dst = ei[e_count + e];           // col (destination)
  float coef = dinv[src] * dinv[dst];
  float4 hv = ((const float4*)(h + (long long)src * NHID))[c];
  float* o = out + (long long)dst * NHID + c * 4;
  unsafeAtomicAdd(o + 0, hv.x * coef);
  unsafeAtomicAdd(o + 1, hv.y * coef);
  unsafeAtomicAdd(o + 2, hv.z * coef);
  unsafeAtomicAdd(o + 3, hv.w * coef);
}

// -------------------- 6. self-loop + bias + PReLU --------------------
__global__ void finalize_kernel(float* __restrict__ out, const float* __restrict__ h,
                                const float* __restrict__ dinv, const float* __restrict__ b,
                                const float* __restrict__ prelu_a, long long n) {
  long long idx = (long long)blockIdx.x * blockDim.x + threadIdx.x;
  if (idx >= n) return;
  int i = (int)(idx >> 7);
  int j = (int)(idx & 127);
  float d = dinv[i];
  float val = out[idx] + h[idx] * d * d + b[j];  // self-loop norm = dinv^2
  float a = prelu_a[0];
  out[idx] = val > 0.0f ? val : a * val;
}

extern "C" void kernel_launch(void* const* d_in, const int* in_sizes, int n_in,
                              void* d_out, int out_size, void* d_ws, size_t ws_size,
                              hipStream_t stream) {
  const float* x  = (const float*)d_in[0];
  const int*   ei = (const int*)d_in[1];   // [2,E] flat: [0..E)=row, [E..2E)=col
  const float* W  = (const float*)d_in[2];
  const float* b  = (const float*)d_in[3];
  const float* pa = (const float*)d_in[4];
  float* out = (float*)d_out;

  // workspace layout (floats): [0..64) scalars | [64..64+N) dinv | h at 100096
  float* wsf       = (float*)d_ws;
  float* inv_sigma = wsf;
  float* dinv      = wsf + 64;
  float* h         = wsf + 100096;  // 512B-aligned, N*NHID floats

  const long long NE = N_EDGES;
  const long long OUTN = (long long)N_NODES * NHID;  // 12.8M

  power_iter_kernel<<<1, 256, 0, stream>>>(W, inv_sigma);
  fill_ones_kernel<<<(N_NODES + 255) / 256, 256, 0, stream>>>(dinv, N_NODES);
  deg_count_kernel<<<(N_EDGES + 255) / 256, 256, 0, stream>>>(ei + N_EDGES, dinv, N_EDGES);
  rsqrt_kernel<<<(N_NODES + 255) / 256, 256, 0, stream>>>(dinv, N_NODES);
  zero_out_kernel<<<(int)((OUTN + 255) / 256), 256, 0, stream>>>(out, OUTN);
  gemm_wmma_kernel<<<N_NODES / 16, 256, 0, stream>>>(x, W, inv_sigma, h);
  edge_scatter_kernel<<<(int)((NE * 32 + 255) / 256), 256, 0, stream>>>(ei, h, dinv, out, NE);
  finalize_kernel<<<(int)((OUTN + 255) / 256), 256, 0, stream>>>(out, h, dinv, b, pa, OUTN);
}